// TransformerMeantimeBlock_60765197304459
// MI455X (gfx1250) — compile-verified
//
#include <hip/hip_runtime.h>
#include <math.h>

typedef float v2f __attribute__((ext_vector_type(2)));
typedef float v4f __attribute__((ext_vector_type(4)));
typedef float v8f __attribute__((ext_vector_type(8)));
typedef int   v4i __attribute__((ext_vector_type(4)));

constexpr int kB = 8, kT = 128, kD = 512, kLa = 2, kLr = 2, kH = 4, kDK = 128, kDFF = 2048;
constexpr int kN = kB * kT;  // 1024 token rows

__device__ __forceinline__ float wave_sum(float v) {
#pragma unroll
  for (int o = 16; o > 0; o >>= 1) v += __shfl_xor(v, o, 32);
  return v;
}
__device__ __forceinline__ float wave_max(float v) {
#pragma unroll
  for (int o = 16; o > 0; o >>= 1) v = fmaxf(v, __shfl_xor(v, o, 32));
  return v;
}

// ---------------- LayerNorm: one block (128 thr) per row of 512 ----------------
__global__ __launch_bounds__(128) void ln_kernel(const float* __restrict__ X,
                                                 const float* __restrict__ g,
                                                 const float* __restrict__ b,
                                                 float* __restrict__ out) {
  const int row = blockIdx.x, tid = threadIdx.x;
  const v4f v = *(const v4f*)(X + (size_t)row * kD + tid * 4);
  float s  = v.x + v.y + v.z + v.w;
  float ss = v.x * v.x + v.y * v.y + v.z * v.z + v.w * v.w;
  s = wave_sum(s);
  ss = wave_sum(ss);
  __shared__ float sm[8];
  const int wave = tid >> 5, lane = tid & 31;
  if (lane == 0) { sm[wave] = s; sm[4 + wave] = ss; }
  __syncthreads();
  const float tot  = sm[0] + sm[1] + sm[2] + sm[3];
  const float tot2 = sm[4] + sm[5] + sm[6] + sm[7];
  const float mu = tot * (1.0f / kD);
  const float rstd = rsqrtf(tot2 * (1.0f / kD) - mu * mu + 1e-5f);
  const v4f gg = *(const v4f*)(g + tid * 4);
  const v4f bb = *(const v4f*)(b + tid * 4);
  v4f o;
  o.x = (v.x - mu) * rstd * gg.x + bb.x;
  o.y = (v.y - mu) * rstd * gg.y + bb.y;
  o.z = (v.z - mu) * rstd * gg.z + bb.z;
  o.w = (v.w - mu) * rstd * gg.w + bb.w;
  *(v4f*)(out + (size_t)row * kD + tid * 4) = o;
}

// ---------------- fp32 WMMA GEMM: C = A @ B(^T) + bias + Res, opt. GELU ----------------
// Wave computes a 16x64 tile (4x v8f accumulators) via V_WMMA_F32_16X16X4_F32.
// A-frag: lane = m + 16*(k>=2), vgpr = k&1.  B-frag mirrored (n in place of m).
// C/D-frag: vgpr r -> m = r + 8*(lane>=16), n = lane&15.
template <bool TRANSB>
__global__ __launch_bounds__(128) void gemm_wmma(
    const float* __restrict__ A, int lda, long long sA,
    const float* __restrict__ Bm, int ldb, long long sB,
    const float* __restrict__ bias, int sBias,
    const float* __restrict__ Res,
    float* __restrict__ C, int ldc, long long sC,
    int M, int N, int K, int act) {
  const int lane = threadIdx.x & 31;
  const int wave = threadIdx.x >> 5;
  const int g = lane >> 4;
  const int r = lane & 15;
  const int n0 = blockIdx.x * 64;
  const int m0 = (blockIdx.y * 4 + wave) * 16;
  const int z = blockIdx.z;
  if (m0 >= M || n0 >= N) return;
  A += (long long)z * sA;
  Bm += (long long)z * sB;
  C += (long long)z * sC;
  const float* brow = bias ? (bias + (long long)z * sBias) : nullptr;

  v8f acc[4] = {};
  const float* Ap = A + (size_t)(m0 + r) * lda + 2 * g;
  for (int k = 0; k < K; k += 4) {
    const v2f a = *(const v2f*)(Ap + k);
#pragma unroll
    for (int j = 0; j < 4; ++j) {
      const int n = n0 + j * 16 + r;
      v2f bv;
      if (TRANSB) {
        bv = *(const v2f*)(Bm + (size_t)n * ldb + k + 2 * g);
      } else {
        const float* bp = Bm + (size_t)(k + 2 * g) * ldb + n;
        bv.x = bp[0];
        bv.y = bp[ldb];
      }
      acc[j] = __builtin_amdgcn_wmma_f32_16x16x4_f32(false, a, false, bv,
                                                     (short)0, acc[j], false, false);
    }
  }
#pragma unroll
  for (int j = 0; j < 4; ++j) {
    const int n = n0 + j * 16 + r;
    const float bvs = brow ? brow[n] : 0.0f;
#pragma unroll
    for (int rr = 0; rr < 8; ++rr) {
      const int m = m0 + rr + 8 * g;
      float v = acc[j][rr] + bvs;
      if (Res) v += Res[(size_t)m * ldc + n];
      if (act) {
        const float c0 = 0.7978845608028654f;  // sqrt(2/pi)
        v = 0.5f * v * (1.0f + tanhf(c0 * (v + 0.044715f * v * v * v)));
      }
      C[(size_t)m * ldc + n] = v;
    }
  }
}

// ---------------- elementwise: qa = q[:, :La]+Pq, ka = k[:, :La]+Pk, Xqb = q[:, La:]+rel_bias ----------------
__global__ __launch_bounds__(256) void prep_qk(const float* __restrict__ qkv,
                                               const float* __restrict__ Pq,
                                               const float* __restrict__ Pk,
                                               const float* __restrict__ rel_bias,
                                               float* __restrict__ qa,
                                               float* __restrict__ ka,
                                               float* __restrict__ Xqb) {
  const int idx = blockIdx.x * 256 + threadIdx.x;  // over (b,l,i,e) = 8*2*128*128
  const int e = idx & (kDK - 1);
  const int i = (idx >> 7) & (kT - 1);
  const int l = (idx >> 14) & 1;
  const int b = idx >> 15;
  const size_t qrow = ((size_t)b * kT + i) * kD;
  const float* q  = qkv;
  const float* kk = qkv + (size_t)kN * kD;
  const size_t pidx = (((size_t)l * kB + b) * kT + i) * kDK + e;
  qa[idx]  = q[qrow + l * kDK + e] + Pq[pidx];
  ka[idx]  = kk[qrow + l * kDK + e] + Pk[pidx];
  Xqb[idx] = q[qrow + (kLa + l) * kDK + e] + rel_bias[l * kDK + e];
}

// ---------------- dominant streaming kernel: scores += rel_kernel . u  (536 MB, NT loads) ----------------
__global__ __launch_bounds__(256) void rel_term(const float* __restrict__ relk,
                                                const float* __restrict__ U,
                                                const float* __restrict__ Xqb,
                                                const float* __restrict__ brk,
                                                float* __restrict__ S) {
  const int i = blockIdx.x, b = blockIdx.y, l = blockIdx.z;
  const int lane = threadIdx.x & 31, wave = threadIdx.x >> 5;
  const float* u = U + (((size_t)b * kLr + l) * kT + i) * kD;
  v4f ur[4];
#pragma unroll
  for (int c = 0; c < 4; ++c) ur[c] = *(const v4f*)(u + c * 128 + lane * 4);
  // constant term per (b,l,i): dot(Xq+rel_bias, brk_l), added to every j
  const float* xq = Xqb + (((size_t)b * kLr + l) * kT + i) * kDK;
  const v4f xv = *(const v4f*)(xq + lane * 4);
  const v4f bv = *(const v4f*)(brk + l * kDK + lane * 4);
  const float cadd = wave_sum(xv.x * bv.x + xv.y * bv.y + xv.z * bv.z + xv.w * bv.w);
  const float* base = relk + ((((size_t)l * kB + b) * kT + i) * kT) * kD;
  float* srow = S + (((size_t)(b * kH + kLa + l)) * kT + i) * kT;
  for (int j = wave; j < kT; j += 8) {
    const float* p = base + (size_t)j * kD;
    float acc = 0.0f;
#pragma unroll
    for (int c = 0; c < 4; ++c) {
      const v4f rv = __builtin_nontemporal_load((const v4f*)(p + c * 128 + lane * 4));
      acc += rv.x * ur[c].x + rv.y * ur[c].y + rv.z * ur[c].z + rv.w * ur[c].w;
    }
    acc = wave_sum(acc);
    if (lane == 0) srow[j] += acc + cadd;
  }
}

// ---------------- scale + mask + softmax over j (one wave per row of 128) ----------------
__global__ __launch_bounds__(256) void softmax_mask(float* __restrict__ S,
                                                    const int* __restrict__ mask) {
  const int row = blockIdx.x * 8 + (threadIdx.x >> 5);  // B*H*T rows
  const int lane = threadIdx.x & 31;
  const int i = row & (kT - 1);
  const int b = row >> 9;  // row / (H*T)
  float* srow = S + (size_t)row * kT;
  const int* mrow = mask + ((size_t)b * kT + i) * kT;
  const float scale = 0.08838834764831843f;  // DK^-0.5
  v4f sv = *(v4f*)(srow + lane * 4);
  const v4i mv = *(const v4i*)(mrow + lane * 4);
  float x0 = (mv.x == 0) ? -1e9f : sv.x * scale;
  float x1 = (mv.y == 0) ? -1e9f : sv.y * scale;
  float x2 = (mv.z == 0) ? -1e9f : sv.z * scale;
  float x3 = (mv.w == 0) ? -1e9f : sv.w * scale;
  const float mx = wave_max(fmaxf(fmaxf(x0, x1), fmaxf(x2, x3)));
  x0 = expf(x0 - mx); x1 = expf(x1 - mx); x2 = expf(x2 - mx); x3 = expf(x3 - mx);
  const float inv = 1.0f / wave_sum(x0 + x1 + x2 + x3);
  sv.x = x0 * inv; sv.y = x1 * inv; sv.z = x2 * inv; sv.w = x3 * inv;
  *(v4f*)(srow + lane * 4) = sv;
}

extern "C" void kernel_launch(void* const* d_in, const int* in_sizes, int n_in,
                              void* d_out, int out_size, void* d_ws, size_t ws_size,
                              hipStream_t stream) {
  (void)in_sizes; (void)n_in; (void)out_size; (void)ws_size;
  const float* x        = (const float*)d_in[0];
  const int*   mask     = (const int*)d_in[1];
  const float* abs_k    = (const float*)d_in[2];
  const float* rel_k    = (const float*)d_in[3];
  const float* Wc       = (const float*)d_in[4];
  const float* bc       = (const float*)d_in[5];
  const float* Wpq      = (const float*)d_in[6];
  const float* bpq      = (const float*)d_in[7];
  const float* Wpk      = (const float*)d_in[8];
  const float* bpk      = (const float*)d_in[9];
  const float* Wrk      = (const float*)d_in[10];
  const float* brk      = (const float*)d_in[11];
  const float* rel_bias = (const float*)d_in[12];
  const float* Wo       = (const float*)d_in[13];
  const float* bo       = (const float*)d_in[14];
  const float* W1       = (const float*)d_in[15];
  const float* b1       = (const float*)d_in[16];
  const float* W2       = (const float*)d_in[17];
  const float* b2       = (const float*)d_in[18];
  const float* ln1_g    = (const float*)d_in[19];
  const float* ln1_b    = (const float*)d_in[20];
  const float* ln2_g    = (const float*)d_in[21];
  const float* ln2_b    = (const float*)d_in[22];
  float* out = (float*)d_out;
  float* ws  = (float*)d_ws;

  // workspace layout (floats)
  float* H1  = ws;                 // 524288  : ln1(x)
  float* QKV = ws + 524288;        // 1572864 : (3,B,T,D)
  float* PQ  = ws + 2097152;       // 262144  : (La,B,T,DK)
  float* PK  = ws + 2359296;       // 262144
  float* QA  = ws + 2621440;       // 262144  : (B,La,T,DK) q+Pq
  float* KA  = ws + 2883584;       // 262144  : k+Pk
  float* XQB = ws + 3145728;       // 262144  : Xq+rel_bias
  float* Uw  = ws + 3407872;       // 1048576 : (B,Lr,T,D)
  float* SC  = ws + 4456448;       // 524288  : scores (B,H,T,T)
  float* ATT = ws + 4980736;       // 524288  : attn out (B,T,D)
  float* X1  = ws + 5505024;       // 524288  : residual 1
  float* H2  = ws + 6029312;       // 524288  : ln2(x1)
  float* MID = ws + 6553600;       // 2097152 : FFN hidden (B,T,DFF)

  // 1) LN1
  ln_kernel<<<kN, 128, 0, stream>>>(x, ln1_g, ln1_b, H1);
  // 2) qkv = h @ Wc + bc   (batched over c=0..2)
  gemm_wmma<false><<<dim3(8, 16, 3), 128, 0, stream>>>(
      H1, kD, 0, Wc, kD, (long long)kD * kD, bc, kD, nullptr,
      QKV, kD, (long long)kN * kD, kN, kD, kD, 0);
  // 3) Pq/Pk = abs_kernel @ Wpq/Wpk + b   (batched over l)
  gemm_wmma<false><<<dim3(2, 16, 2), 128, 0, stream>>>(
      abs_k, kD, (long long)kN * kD, Wpq, kDK, (long long)kD * kDK, bpq, kDK, nullptr,
      PQ, kDK, (long long)kN * kDK, kN, kDK, kD, 0);
  gemm_wmma<false><<<dim3(2, 16, 2), 128, 0, stream>>>(
      abs_k, kD, (long long)kN * kD, Wpk, kDK, (long long)kD * kDK, bpk, kDK, nullptr,
      PK, kDK, (long long)kN * kDK, kN, kDK, kD, 0);
  // 4) elementwise prep
  prep_qk<<<kB * kLa * kT * kDK / 256, 256, 0, stream>>>(QKV, PQ, PK, rel_bias, QA, KA, XQB);
  // 5) per-layer score GEMMs (NT) and U = (Xq+rb) @ Wrk^T (NT)
  for (int l = 0; l < kLa; ++l) {
    // abs scores: (q+Pq) @ (k+Pk)^T -> SC[:, l]
    gemm_wmma<true><<<dim3(2, 2, 8), 128, 0, stream>>>(
        QA + l * kT * kDK, kDK, (long long)kLa * kT * kDK,
        KA + l * kT * kDK, kDK, (long long)kLa * kT * kDK, nullptr, 0, nullptr,
        SC + l * kT * kT, kT, (long long)kH * kT * kT, kT, kT, kDK, 0);
    // rel base: Xq @ k^T -> SC[:, La+l]
    gemm_wmma<true><<<dim3(2, 2, 8), 128, 0, stream>>>(
        QKV + (kLa + l) * kDK, kD, (long long)kT * kD,
        QKV + (size_t)kN * kD + (kLa + l) * kDK, kD, (long long)kT * kD, nullptr, 0, nullptr,
        SC + (kLa + l) * kT * kT, kT, (long long)kH * kT * kT, kT, kT, kDK, 0);
    // U: (T,DK) @ Wrk[l](D,DK)^T -> (T,D)
    gemm_wmma<true><<<dim3(8, 2, 8), 128, 0, stream>>>(
        XQB + l * kT * kDK, kDK, (long long)kLr * kT * kDK,
        Wrk + (size_t)l * kD * kDK, kDK, 0, nullptr, 0, nullptr,
        Uw + l * kT * kD, kD, (long long)kLr * kT * kD, kT, kD, kDK, 0);
  }
  // 6) dominant 536MB stream: scores += rel_kernel . u  (+ Xqb.brk const)
  rel_term<<<dim3(kT, kB, kLr), 256, 0, stream>>>(rel_k, Uw, XQB, brk, SC);
  // 7) scale + mask + softmax
  softmax_mask<<<kB * kH * kT / 8, 256, 0, stream>>>(SC, mask);
  // 8) out = p @ v  (per head, batched over b)
  for (int hh = 0; hh < kH; ++hh) {
    gemm_wmma<false><<<dim3(2, 2, 8), 128, 0, stream>>>(
        SC + hh * kT * kT, kT, (long long)kH * kT * kT,
        QKV + 2LL * kN * kD + hh * kDK, kD, (long long)kT * kD, nullptr, 0, nullptr,
        ATT + hh * kDK, kD, (long long)kT * kD, kT, kDK, kT, 0);
  }
  // 9) x1 = x + att @ Wo + bo
  gemm_wmma<false><<<dim3(8, 16, 1), 128, 0, stream>>>(
      ATT, kD, 0, Wo, kD, 0, bo, 0, x, X1, kD, 0, kN, kD, kD, 0);
  // 10) LN2
  ln_kernel<<<kN, 128, 0, stream>>>(X1, ln2_g, ln2_b, H2);
  // 11) FFN1 with fused GELU
  gemm_wmma<false><<<dim3(32, 16, 1), 128, 0, stream>>>(
      H2, kD, 0, W1, kDFF, 0, b1, 0, nullptr, MID, kDFF, 0, kN, kDFF, kD, 1);
  // 12) FFN2 + residual -> d_out
  gemm_wmma<false><<<dim3(8, 16, 1), 128, 0, stream>>>(
      MID, kDFF, 0, W2, kD, 0, b2, 0, X1, out, kD, 0, kN, kD, kDFF, 0);
}